// Test_62964220559567
// MI455X (gfx1250) — compile-verified
//
#include <hip/hip_runtime.h>
#include <hip/hip_bf16.h>

typedef __attribute__((ext_vector_type(2))) float v2f;
typedef __attribute__((ext_vector_type(8))) float v8f;

#define NNODES 262144
#define NEDGES 2097152
#define BN_EPS 1e-5f

// Monotone bijection float -> u32: preserves order under unsigned compare.
// key(-inf)=0x007FFFFF, key(+inf)=0xFF800000; 0 is below every real key,
// so a zero-initialized buffer encodes "no value yet" (segment empty).
__device__ __forceinline__ unsigned floatToKey(float f) {
    unsigned b = __float_as_uint(f);
    return b ^ (unsigned)(((int)b >> 31) | 0x80000000);
}
__device__ __forceinline__ float keyToFloat(unsigned u) {
    return __uint_as_float(u & 0x80000000u ? (u ^ 0x80000000u) : ~u);
}

__global__ void fill4_kernel(uint4* p, int n4, unsigned v) {
    int i = blockIdx.x * blockDim.x + threadIdx.x;
    if (i < n4) p[i] = make_uint4(v, v, v, v);
}

// conv1: msg = [x_j, pos_j - pos_i] @ W1 + b1 ; scatter segment-max over dst.
// One wave handles 16 edges with one V_WMMA_F32_16X16X4_F32:
//   A = 16x4 (edges x feats), B = 4x16 (feats x channels), D = 16x16 f32.
__global__ __launch_bounds__(256) void conv1_kernel(
    const float* __restrict__ x, const float* __restrict__ pos,
    const int* __restrict__ ei, const float* __restrict__ W1,
    const float* __restrict__ b1, unsigned* __restrict__ agg1) {
    const int lane = threadIdx.x & 31;
    const int tile = blockIdx.x * 8 + (threadIdx.x >> 5);
    const int e0   = tile * 16;
    const int half = lane >> 4;   // 0: K=0,1 ; 1: K=2,3 (A) / rows (B)
    const int n    = lane & 15;   // channel (B/C/D column), also A row M

    // dst of the 8 edges this lane will scatter for (M = r + 8*half);
    // issued early as one load clause, broadcast within each half.
    int dsts[8];
    #pragma unroll
    for (int r = 0; r < 8; ++r) dsts[r] = ei[NEDGES + e0 + 8*half + r];

    const int src = ei[e0 + n];
    const int dst = ei[NEDGES + e0 + n];
    const float rx = pos[src*3+0] - pos[dst*3+0];
    const float ry = pos[src*3+1] - pos[dst*3+1];
    const float rz = pos[src*3+2] - pos[dst*3+2];
    // A 16x4 f32 layout: lanes 0-15 -> {K0,K1}, lanes 16-31 -> {K2,K3}
    v2f a;
    if (half == 0) { a.x = x[src]; a.y = rx; }
    else           { a.x = ry;     a.y = rz; }
    // B 4x16 f32 layout: VGPR r holds row K=r (lanes 0-15), K=r+2 (lanes 16-31)
    v2f b;
    b.x = W1[(2*half + 0)*16 + n];
    b.y = W1[(2*half + 1)*16 + n];
    v8f c = {};
    c = __builtin_amdgcn_wmma_f32_16x16x4_f32(false, a, false, b, (short)0, c,
                                              false, false);
    const float bias = b1[n];
    // C/D: VGPR r holds edge M=r (lanes 0-15) / M=r+8 (lanes 16-31), channel n
    #pragma unroll
    for (int r = 0; r < 8; ++r) {
        unsigned idx = (unsigned)dsts[r] * 16u + (unsigned)n;
        atomicMax(&agg1[idx], floatToKey(c[r] + bias));
    }
}

// conv2: msg = [h1_j(16), rel(3), 0pad] @ W2(20x32 padded) + b2 ; scatter-max.
// Per 16-edge tile: 5 K-chunks x 2 N-tiles = 10 chained f32 WMMAs.
__global__ __launch_bounds__(256) void conv2_kernel(
    const float* __restrict__ pos, const int* __restrict__ ei,
    const float* __restrict__ h1, const float* __restrict__ W2,
    const float* __restrict__ b2, unsigned* __restrict__ agg2) {
    const int lane = threadIdx.x & 31;
    const int tile = blockIdx.x * 8 + (threadIdx.x >> 5);
    const int e0   = tile * 16;
    const int half = lane >> 4;
    const int n    = lane & 15;

    int dsts[8];
    #pragma unroll
    for (int r = 0; r < 8; ++r) dsts[r] = ei[NEDGES + e0 + 8*half + r];

    const int src = ei[e0 + n];
    const int dst = ei[NEDGES + e0 + n];
    const float rx = pos[src*3+0] - pos[dst*3+0];
    const float ry = pos[src*3+1] - pos[dst*3+1];
    const float rz = pos[src*3+2] - pos[dst*3+2];
    // A feats: [h1[0..15], rx, ry, rz, 0]; chunk kc covers K=4kc..4kc+3
    v2f a[5];
    const float* hrow = h1 + (size_t)src * 16;
    #pragma unroll
    for (int kc = 0; kc < 4; ++kc) {
        int j = 4*kc + 2*half;
        a[kc].x = hrow[j];
        a[kc].y = hrow[j + 1];
    }
    if (half == 0) { a[4].x = rx; a[4].y = ry;   }
    else           { a[4].x = rz; a[4].y = 0.0f; }

    v8f c0 = {}, c1 = {};
    #pragma unroll
    for (int kc = 0; kc < 5; ++kc) {
        int k0 = 4*kc + 2*half;
        int k1 = k0 + 1;
        v2f bA, bB;   // N-tile 0 (ch 0-15) and N-tile 1 (ch 16-31)
        bA.x = (k0 < 19) ? W2[k0*32 + n]      : 0.0f;
        bA.y = (k1 < 19) ? W2[k1*32 + n]      : 0.0f;
        bB.x = (k0 < 19) ? W2[k0*32 + 16 + n] : 0.0f;
        bB.y = (k1 < 19) ? W2[k1*32 + 16 + n] : 0.0f;
        c0 = __builtin_amdgcn_wmma_f32_16x16x4_f32(false, a[kc], false, bA,
                                                   (short)0, c0, false, false);
        c1 = __builtin_amdgcn_wmma_f32_16x16x4_f32(false, a[kc], false, bB,
                                                   (short)0, c1, false, false);
    }
    const float bias0 = b2[n];
    const float bias1 = b2[16 + n];
    #pragma unroll
    for (int r = 0; r < 8; ++r) {
        unsigned base = (unsigned)dsts[r] * 32u + (unsigned)n;
        atomicMax(&agg2[base],       floatToKey(c0[r] + bias0));
        atomicMax(&agg2[base + 16u], floatToKey(c1[r] + bias1));
    }
}

// BN(eval) + ReLU for the 16-channel layer; key==0 (no in-edges) -> 0.
__global__ void bn_relu_16(const unsigned* __restrict__ agg,
                           const float* __restrict__ mean,
                           const float* __restrict__ var,
                           const float* __restrict__ gamma,
                           const float* __restrict__ beta,
                           float* __restrict__ h) {
    int i = blockIdx.x * blockDim.x + threadIdx.x;   // exact grid, N*16 threads
    int ch = i & 15;
    unsigned k = agg[i];
    float v = (k == 0u) ? 0.0f : keyToFloat(k);
    float y = (v - mean[ch]) * rsqrtf(var[ch] + BN_EPS) * gamma[ch] + beta[ch];
    h[i] = fmaxf(y, 0.0f);
}

// Fused BN2 + ReLU + 8x8 voxel max-pool. LDS-staged [64][32] max per block;
// ReLU output >= 0 so 0-init + int atomicMax on float bits is order-correct.
__global__ __launch_bounds__(256) void bn2_pool_kernel(
    const unsigned* __restrict__ agg2, const float* __restrict__ pos,
    const float* __restrict__ mean, const float* __restrict__ var,
    const float* __restrict__ gamma, const float* __restrict__ beta,
    float* __restrict__ out) {
    __shared__ float smax[64 * 32];
    for (int i = threadIdx.x; i < 64 * 32; i += blockDim.x) smax[i] = 0.0f;
    __syncthreads();
    for (int node = blockIdx.x * blockDim.x + threadIdx.x; node < NNODES;
         node += gridDim.x * blockDim.x) {
        float px = pos[node*3 + 0];
        float py = pos[node*3 + 1];
        int gx = (int)floorf(px * (1.0f / 16.0f));
        int gy = (int)floorf(py * (1.0f / 16.0f));
        int cl = gx + gy * 8;
        cl = min(max(cl, 0), 63);
        #pragma unroll
        for (int ch = 0; ch < 32; ++ch) {
            unsigned k = agg2[(size_t)node * 32 + ch];
            float v = (k == 0u) ? 0.0f : keyToFloat(k);
            float y = (v - mean[ch]) * rsqrtf(var[ch] + BN_EPS) * gamma[ch] + beta[ch];
            y = fmaxf(y, 0.0f);
            atomicMax((int*)&smax[cl * 32 + ch], __float_as_int(y));
        }
    }
    __syncthreads();
    for (int i = threadIdx.x; i < 64 * 32; i += blockDim.x)
        atomicMax((int*)&out[i], __float_as_int(smax[i]));
}

extern "C" void kernel_launch(void* const* d_in, const int* in_sizes, int n_in,
                              void* d_out, int out_size, void* d_ws, size_t ws_size,
                              hipStream_t stream) {
    const float* x   = (const float*)d_in[0];
    const float* pos = (const float*)d_in[1];
    const int*   ei  = (const int*)d_in[2];
    const float* W1  = (const float*)d_in[3];
    const float* b1  = (const float*)d_in[4];
    const float* m1  = (const float*)d_in[5];
    const float* v1  = (const float*)d_in[6];
    const float* g1  = (const float*)d_in[7];
    const float* be1 = (const float*)d_in[8];
    const float* W2  = (const float*)d_in[9];
    const float* b2  = (const float*)d_in[10];
    const float* m2  = (const float*)d_in[11];
    const float* v2  = (const float*)d_in[12];
    const float* g2  = (const float*)d_in[13];
    const float* be2 = (const float*)d_in[14];
    float* out = (float*)d_out;

    // Workspace: h1 [N*16] f32 (16MB) then key-encoded agg region [N*32] u32
    // (32MB, reused for agg1 then agg2). Total 48MB.
    float*    h1  = (float*)d_ws;
    unsigned* agg = (unsigned*)(h1 + (size_t)NNODES * 16);

    const int threads = 256;
    // init agg1 keys = 0 ("empty"), out = 0.0f
    fill4_kernel<<<(NNODES*16/4 + threads-1)/threads, threads, 0, stream>>>(
        (uint4*)agg, NNODES*16/4, 0u);
    fill4_kernel<<<(2048/4 + threads-1)/threads, threads, 0, stream>>>(
        (uint4*)out, 2048/4, 0u);
    // conv1: 16 edges per wave, 8 waves per block
    conv1_kernel<<<NEDGES/(16*8), threads, 0, stream>>>(x, pos, ei, W1, b1, agg);
    bn_relu_16<<<NNODES*16/threads, threads, 0, stream>>>(agg, m1, v1, g1, be1, h1);
    // init agg2 keys = 0 (reuses agg region after bn1 consumed agg1)
    fill4_kernel<<<(NNODES*32/4 + threads-1)/threads, threads, 0, stream>>>(
        (uint4*)agg, NNODES*32/4, 0u);
    conv2_kernel<<<NEDGES/(16*8), threads, 0, stream>>>(pos, ei, h1, W2, b2, agg);
    bn2_pool_kernel<<<256, threads, 0, stream>>>(agg, pos, m2, v2, g2, be2, out);
}